// IE_22600117911987
// MI455X (gfx1250) — compile-verified
//
#include <hip/hip_runtime.h>
#include <hip/hip_bf16.h>

// CDNA5 / gfx1250, wave32. Uses V_WMMA_F32_16X16X4_F32 for both the
// embedding-projection GEMM and the per-word 16x16x128 QK^T.

typedef __attribute__((ext_vector_type(2))) float v2f;
typedef __attribute__((ext_vector_type(4))) float v4f;
typedef __attribute__((ext_vector_type(8))) float v8f;

#define VOCAB 4096
#define DMODEL 128
#define HEADS 8
#define LWORD 16
#define NWORDS 16384
#define QKV3 (3 * HEADS * DMODEL)   // 3072
#define PADC 0

// ---------------------------------------------------------------------------
// Kernel 1: pqkv = sinusoid_pos(16,128) @ pos_qkv_w(128,3072) -> (16,3072)
// ---------------------------------------------------------------------------
__global__ __launch_bounds__(256) void IE_pqkv_kernel(
    const float* __restrict__ wq, float* __restrict__ pqkv)
{
  __shared__ float pos[LWORD * DMODEL];     // 8 KB
  const int tid = threadIdx.x;
  for (int i = tid; i < LWORD * DMODEL; i += 256) {
    int p = i >> 7, d = i & 127, f = d & 63;
    // angle = p * 10000^(-f/64)
    float ang = (float)p * __expf(-(float)f * (9.210340371976184f / 64.0f));
    pos[i] = (d < 64) ? __cosf(ang) : __sinf(ang);
  }
  __syncthreads();
  int j = blockIdx.x * 256 + tid;           // 0 .. 16*3072-1 (grid sized exactly)
  int p = j / QKV3, col = j % QKV3;
  float acc = 0.0f;
  #pragma unroll 8
  for (int d = 0; d < DMODEL; ++d)
    acc = fmaf(pos[p * DMODEL + d], wq[(size_t)d * QKV3 + col], acc);
  pqkv[j] = acc;
}

// ---------------------------------------------------------------------------
// Kernel 2: cqkv = emb(4096,128) @ char_qkv_w(128,3072) -> (4096,3072)
// One 16x16 output tile per wave; 32 x V_WMMA_F32_16X16X4_F32 over K=128.
// f32 frag layout (ISA 7.12.2): lane L -> row/col L%16, K = 2*(L/16)+{0,1}.
// ---------------------------------------------------------------------------
__global__ __launch_bounds__(256) void IE_cqkv_gemm(
    const float* __restrict__ emb,
    const float* __restrict__ wq,
    float* __restrict__ outm)
{
  const int tid  = threadIdx.x;
  const int wave = tid >> 5;
  const int lane = tid & 31;
  const int tile = blockIdx.x * 8 + wave;         // 256*192 tiles total
  const int tm   = tile / (QKV3 / 16);            // 0..255
  const int tn   = tile % (QKV3 / 16);            // 0..191
  const int n    = lane & 15;
  const int half = lane >> 4;

  const float* arow = emb + (size_t)(tm * 16 + n) * DMODEL;
  const float* bcol = wq + tn * 16 + n;

  v8f acc = {};
  #pragma unroll 8
  for (int kk = 0; kk < 32; ++kk) {
    int dd = kk * 4 + half * 2;
    v2f a = *(const v2f*)(arow + dd);                          // A[m, dd..dd+1]
    v2f b;
    b.x = bcol[(size_t)dd * QKV3];                             // B[dd,   n]
    b.y = bcol[(size_t)(dd + 1) * QKV3];                       // B[dd+1, n]
    acc = __builtin_amdgcn_wmma_f32_16x16x4_f32(
        false, a, false, b, (short)0, acc, false, false);
  }
  // C layout: VGPR r, lane L -> row r + 8*(L/16), col L%16
  float* orow = outm + (size_t)(tm * 16 + 8 * half) * QKV3 + tn * 16 + n;
  #pragma unroll
  for (int r = 0; r < 8; ++r) orow[(size_t)r * QKV3] = acc[r];
}

// ---------------------------------------------------------------------------
// Kernel 3: per-word attention + pooling + LayerNorm + head-mixing MLP.
// 1 block = 1 word; 8 waves = 8 heads. cqkv (48 MB) is L2-resident.
// ---------------------------------------------------------------------------
__global__ __launch_bounds__(256) void IE_attn_kernel(
    const int*   __restrict__ char_code,   // [W][16]
    const float* __restrict__ cqkv,        // [4096][3][8][128]
    const float* __restrict__ pqkv,        // [16][3][8][128]
    const float* __restrict__ ln_g,
    const float* __restrict__ ln_b,
    const float* __restrict__ w1,          // [8][16]
    const float* __restrict__ b1,          // [16]
    const float* __restrict__ w2,          // [16][4]
    const float* __restrict__ b2,          // [4]
    float*       __restrict__ outp)        // [W][512]
{
  __shared__ float sh_out[HEADS][DMODEL];  // 4 KB: post-LN per-head vectors
  __shared__ float sh_w1[8][16];
  __shared__ float sh_b1[16];
  __shared__ float sh_w2[16][4];
  __shared__ float sh_b2[4];

  const int w    = blockIdx.x;
  const int tid  = threadIdx.x;
  const int h    = tid >> 5;               // wave id == head
  const int lane = tid & 31;
  const int n    = lane & 15;              // char index (A row / B col)
  const int half = lane >> 4;

  // stage tiny MLP weights into LDS
  if (tid < 128)                    sh_w1[tid >> 4][tid & 15] = w1[tid];
  else if (tid < 144)               sh_b1[tid - 128] = b1[tid - 128];
  else if (tid >= 160 && tid < 224) sh_w2[(tid - 160) >> 2][(tid - 160) & 3] = w2[tid - 160];
  else if (tid >= 224 && tid < 228) sh_b2[tid - 224] = b2[tid - 224];

  const int myc = char_code[w * LWORD + n];
  const unsigned vm = (unsigned)__ballot(myc != PADC);   // bits 0..15 = valid(l)
  const float count = (float)__popc(vm & 0xFFFFu);

  // Q[l] = char_q[c_l] + pos_q[l] ; K[m] = char_k[c_m] + pos_k[m]
  const float* cq = cqkv + (size_t)myc * QKV3 +        0 + h * DMODEL;
  const float* ck = cqkv + (size_t)myc * QKV3 + 1 * 1024 + h * DMODEL;
  const float* pq = pqkv + (size_t)n   * QKV3 +        0 + h * DMODEL;
  const float* pk = pqkv + (size_t)n   * QKV3 + 1 * 1024 + h * DMODEL;

  v8f acc = {};
  #pragma unroll
  for (int kk = 0; kk < 32; ++kk) {
    int dd = kk * 4 + half * 2;
    v2f a = *(const v2f*)(cq + dd) + *(const v2f*)(pq + dd);
    v2f b = *(const v2f*)(ck + dd) + *(const v2f*)(pk + dd);
    acc = __builtin_amdgcn_wmma_f32_16x16x4_f32(
        false, a, false, b, (short)0, acc, false, false);
  }

  // PAD key mask (column n == my char) + 1/sqrt(128) scale
  const float maskadd = (myc == PADC) ? -1.0e9f : 0.0f;
  const float invs = 0.08838834764831845f;
  #pragma unroll
  for (int r = 0; r < 8; ++r) acc[r] = (acc[r] + maskadd) * invs;

  // softmax over keys (16 lanes of each half hold one row), then masked pool
  float pooled = 0.0f;
  #pragma unroll
  for (int r = 0; r < 8; ++r) {
    float s = acc[r];
    float mx = s;
    mx = fmaxf(mx, __shfl_xor(mx, 1, 32));
    mx = fmaxf(mx, __shfl_xor(mx, 2, 32));
    mx = fmaxf(mx, __shfl_xor(mx, 4, 32));
    mx = fmaxf(mx, __shfl_xor(mx, 8, 32));
    float e = __expf(s - mx);
    float su = e;
    su += __shfl_xor(su, 1, 32);
    su += __shfl_xor(su, 2, 32);
    su += __shfl_xor(su, 4, 32);
    su += __shfl_xor(su, 8, 32);
    float a = e / su;
    int l = r + 8 * half;                              // query row index
    pooled += ((vm >> l) & 1u) ? a : 0.0f;
  }
  pooled += __shfl_xor(pooled, 16, 32);                // combine row halves
  pooled /= count;                                     // lanes n and n+16: pooled[n]

  // out[d] = sum_m pooled[m] * (char_v[c_m,h,d] + pos_v[m,h,d]); 4 d per lane
  const int dbase = lane * 4;                          // 0..124
  v4f o = {0.f, 0.f, 0.f, 0.f};
  #pragma unroll
  for (int m = 0; m < 16; ++m) {
    float pm = __shfl(pooled, m, 32);
    int   cm = __shfl(myc,    m, 32);
    v4f vv = *(const v4f*)(cqkv + (size_t)cm * QKV3 + 2 * 1024 + h * DMODEL + dbase)
           + *(const v4f*)(pqkv + (size_t)m  * QKV3 + 2 * 1024 + h * DMODEL + dbase);
    o += pm * vv;
  }

  // LayerNorm over D=128 (wave-wide reduction)
  float ssum = o.x + o.y + o.z + o.w;
  #pragma unroll
  for (int off = 16; off >= 1; off >>= 1) ssum += __shfl_xor(ssum, off, 32);
  float mu = ssum * (1.0f / 128.0f);
  v4f c = o - mu;
  float vsum = c.x * c.x + c.y * c.y + c.z * c.z + c.w * c.w;
  #pragma unroll
  for (int off = 16; off >= 1; off >>= 1) vsum += __shfl_xor(vsum, off, 32);
  float inv = rsqrtf(vsum * (1.0f / 128.0f) + 1e-5f);
  v4f g  = *(const v4f*)(ln_g + dbase);
  v4f bb = *(const v4f*)(ln_b + dbase);
  v4f y  = c * inv * g + bb;
  *(v4f*)&sh_out[h][dbase] = y;
  __syncthreads();

  // head-mixing MLP: per d, 8 -> 16 (relu) -> 4
  if (tid < DMODEL) {
    const int d = tid;
    float xh[8];
    #pragma unroll
    for (int hh = 0; hh < 8; ++hh) xh[hh] = sh_out[hh][d];
    float y4[4] = {sh_b2[0], sh_b2[1], sh_b2[2], sh_b2[3]};
    #pragma unroll
    for (int j = 0; j < 16; ++j) {
      float t = sh_b1[j];
      #pragma unroll
      for (int hh = 0; hh < 8; ++hh) t = fmaf(xh[hh], sh_w1[hh][j], t);
      t = fmaxf(t, 0.0f);
      #pragma unroll
      for (int k = 0; k < 4; ++k) y4[k] = fmaf(t, sh_w2[j][k], y4[k]);
    }
    v4f yo = {y4[0], y4[1], y4[2], y4[3]};
    *(v4f*)(outp + (size_t)w * 512 + d * 4) = yo;
  }
}

// ---------------------------------------------------------------------------
extern "C" void kernel_launch(void* const* d_in, const int* in_sizes, int n_in,
                              void* d_out, int out_size, void* d_ws, size_t ws_size,
                              hipStream_t stream) {
  const int*   char_code  = (const int*)  d_in[0];
  const float* emb        = (const float*)d_in[1];
  const float* char_qkv_w = (const float*)d_in[2];
  const float* pos_qkv_w  = (const float*)d_in[3];
  const float* ln_g       = (const float*)d_in[4];
  const float* ln_b       = (const float*)d_in[5];
  const float* mlp_w1     = (const float*)d_in[6];
  const float* mlp_b1     = (const float*)d_in[7];
  const float* mlp_w2     = (const float*)d_in[8];
  const float* mlp_b2     = (const float*)d_in[9];
  float* outp = (float*)d_out;

  float* cqkv = (float*)d_ws;                         // 4096*3072 f32 = 48 MB
  float* pqkv = cqkv + (size_t)VOCAB * QKV3;          // 16*3072 f32

  IE_pqkv_kernel<<<(LWORD * QKV3) / 256, 256, 0, stream>>>(pos_qkv_w, pqkv);
  IE_cqkv_gemm<<<(VOCAB / 16) * (QKV3 / 16) / 8, 256, 0, stream>>>(emb, char_qkv_w, cqkv);
  IE_attn_kernel<<<NWORDS, 256, 0, stream>>>(char_code, cqkv, pqkv, ln_g, ln_b,
                                             mlp_w1, mlp_b1, mlp_w2, mlp_b2, outp);
}